// ExplanationGuidedPrimaryModel_86474871537929
// MI455X (gfx1250) — compile-verified
//
#include <hip/hip_runtime.h>
#include <hip/hip_bf16.h>
#include <math.h>

// Model dims (fixed by reference)
#define Bb 128
#define Nn 64
#define Dd 128
#define EFd 8
#define EHd 256
#define THd 128
#define FUSED_DIM 388          // TH + 2D + 4
#define OUT_STRIDE 4176        // 8+1+3+1+3+64+4096

typedef float v2f __attribute__((ext_vector_type(2)));
typedef float v8f __attribute__((ext_vector_type(8)));

__device__ __forceinline__ float wave_sum32(float v) {
#pragma unroll
  for (int off = 16; off; off >>= 1) v += __shfl_down(v, off);
  return v;
}

__device__ __forceinline__ float sigmoidf_(float x) {
  return 1.0f / (1.0f + expf(-x));
}

// ---------------------------------------------------------------------------
// C[M,N] = act(A[M,K] @ B[K,N] + bias)   via V_WMMA_F32_16X16X4_F32
// One 16x16 output tile per wave32; block = 128 threads = 4 waves (64 cols).
// Requires M%16==0, N%64==0, K%4==0 (true for every call here).
// act: 0 = none, 1 = relu, 2 = tanh
// ---------------------------------------------------------------------------
__global__ __launch_bounds__(128)
void gemm_wmma_f32(const float* __restrict__ A, int lda,
                   const float* __restrict__ Bm, int ldb,
                   const float* __restrict__ bias,
                   float* __restrict__ C, int ldc,
                   int K, int act)
{
  const int wave = threadIdx.x >> 5;
  const int lane = threadIdx.x & 31;
  const int half = lane >> 4;      // which K-pair / which row-half
  const int l16  = lane & 15;
  const int tileM = blockIdx.x * 16;
  const int tileN = (blockIdx.y * 4 + wave) * 16;

  v8f acc = {};
  const float* Arow = A + (size_t)(tileM + l16) * lda;
  const float* Bcol = Bm + tileN + l16;
  for (int kk = 0; kk < K; kk += 4) {
    const int ka = kk + 2 * half;
    v2f a, b;
    // A 16x4 frag: lanes 0-15 rows M, v0=K(2*half), v1=K(2*half+1)
    a.x = Arow[ka];
    a.y = Arow[ka + 1];
    // B 4x16 frag: mirrored on N (B^T has A's layout)
    b.x = Bcol[(size_t)ka * ldb];
    b.y = Bcol[(size_t)(ka + 1) * ldb];
    acc = __builtin_amdgcn_wmma_f32_16x16x4_f32(false, a, false, b,
                                                (short)0, acc, false, false);
  }
  const int col = tileN + l16;
  const float bv = bias ? bias[col] : 0.0f;
#pragma unroll
  for (int r = 0; r < 8; ++r) {
    const int row = tileM + r + 8 * half;   // C/D layout: VGPR r -> M=r / r+8
    float v = acc[r] + bv;
    if (act == 1)      v = fmaxf(v, 0.0f);
    else if (act == 2) v = tanhf(v);
    C[(size_t)row * ldc + col] = v;
  }
}

// ---------------------------------------------------------------------------
// organ_susc[row] = sigmoid(dot(O1[row,:256], Wo2) + bo2) ; row = b*64+n
// 8 waves/block, one row per wave. Also writes output organ section.
// ---------------------------------------------------------------------------
__global__ __launch_bounds__(256)
void organ_kernel(const float* __restrict__ O1, const float* __restrict__ Wo2,
                  const float* __restrict__ bo2,
                  float* __restrict__ organ_s, float* __restrict__ out)
{
  const int row  = blockIdx.x * 8 + (threadIdx.x >> 5);
  const int lane = threadIdx.x & 31;
  const float4* o = (const float4*)(O1 + (size_t)row * EHd);
  const float4* w = (const float4*)Wo2;
  float acc = 0.0f;
#pragma unroll
  for (int c = 0; c < 2; ++c) {
    const int k4 = c * 32 + lane;
    const float4 ov = o[k4], wv = w[k4];
    acc += ov.x * wv.x + ov.y * wv.y + ov.z * wv.z + ov.w * wv.w;
  }
  acc = wave_sum32(acc);
  if (lane == 0) {
    const float s = sigmoidf_(acc + bo2[0]);
    organ_s[row] = s;
    const int b = row >> 6, n = row & 63;
    out[(size_t)b * OUT_STRIDE + 16 + n] = s;
  }
}

// ---------------------------------------------------------------------------
// Attention pool + LN1. One block per batch. 256 threads.
//   scores[n] = dot(S1[b,n,:], Wp2)+bp2 ; softmax over n ; pooled = sum pw*z
//   lnpool[b,:] = LN(pooled)*g1+b1
// ---------------------------------------------------------------------------
__global__ __launch_bounds__(256)
void pool_kernel(const float* __restrict__ z, const float* __restrict__ S1,
                 const float* __restrict__ Wp2, const float* __restrict__ bp2,
                 const float* __restrict__ g1, const float* __restrict__ b1,
                 float* __restrict__ lnpool)
{
  const int b = blockIdx.x;
  const int t = threadIdx.x;
  const int wave = t >> 5, lane = t & 31;
  __shared__ float sc[Nn];
  __shared__ float pool_s[Dd];
  __shared__ float smax_s, ssum_s, mu_s, var_s;

  // scores: each wave does 8 rows
#pragma unroll
  for (int q = 0; q < 8; ++q) {
    const int n = wave * 8 + q;
    const float* s1 = S1 + (size_t)(b * Nn + n) * Dd;
    float a = 0.0f;
    for (int k = lane; k < Dd; k += 32) a += s1[k] * Wp2[k];
    a = wave_sum32(a);
    if (lane == 0) sc[n] = a + bp2[0];
  }
  __syncthreads();
  if (t == 0) {
    float m = sc[0];
    for (int n = 1; n < Nn; ++n) m = fmaxf(m, sc[n]);
    smax_s = m;
  }
  __syncthreads();
  if (t < Nn) sc[t] = expf(sc[t] - smax_s);
  __syncthreads();
  if (t == 0) {
    float s = 0.0f;
    for (int n = 0; n < Nn; ++n) s += sc[n];
    ssum_s = s;
  }
  __syncthreads();
  if (t < Nn) sc[t] /= ssum_s;
  __syncthreads();
  if (t < Dd) {
    float a = 0.0f;
    for (int n = 0; n < Nn; ++n) a += sc[n] * z[(size_t)(b * Nn + n) * Dd + t];
    pool_s[t] = a;
  }
  __syncthreads();
  if (t == 0) {
    float mu = 0.0f;
    for (int d = 0; d < Dd; ++d) mu += pool_s[d];
    mu /= Dd;
    float var = 0.0f;
    for (int d = 0; d < Dd; ++d) { float c = pool_s[d] - mu; var += c * c; }
    var /= Dd;
    mu_s = mu; var_s = var;
  }
  __syncthreads();
  if (t < Dd)
    lnpool[(size_t)b * Dd + t] =
        (pool_s[t] - mu_s) * rsqrtf(var_s + 1e-5f) * g1[t] + b1[t];
}

// ---------------------------------------------------------------------------
// Edge head (fused, never materializes [B,N,N,EH]):
//   edge_prob[b,i,j] = sigmoid(sum_k relu(ZI[b,i,k]+ZJ[b,j,k]+EFW[i,j,k]+be1[k])
//                              * We2[k] + be2) * mask[i,j]
// One block per (b,i); zi+be1 and We2 cached in LDS; one wave per j (8 j/wave).
// float4 (b128) loads throughout; prefetch next j-row (global_prefetch_b8).
// i==0 rows also produce poe (masked primary row).
// ---------------------------------------------------------------------------
__global__ __launch_bounds__(256)
void edge_kernel(const float* __restrict__ ZI, const float* __restrict__ ZJ,
                 const float* __restrict__ EFW, const float* __restrict__ be1,
                 const float* __restrict__ We2, const float* __restrict__ be2,
                 const float* __restrict__ mask,
                 float* __restrict__ out, float* __restrict__ poe)
{
  const int bi = blockIdx.x;
  const int b = bi >> 6;
  const int i = bi & 63;
  __shared__ __align__(16) float zi_s[EHd];
  __shared__ __align__(16) float w2_s[EHd];
  const int t = threadIdx.x;
  zi_s[t] = ZI[(size_t)(b * Nn + i) * EHd + t] + be1[t];
  w2_s[t] = We2[t];
  __syncthreads();
  const int wave = t >> 5, lane = t & 31;
  const float b2 = be2[0];
  const float4* zi4 = (const float4*)zi_s;
  const float4* w24 = (const float4*)w2_s;
#pragma unroll
  for (int jj = 0; jj < 8; ++jj) {
    const int j = wave * 8 + jj;
    const float4* zj = (const float4*)(ZJ + (size_t)(b * Nn + j) * EHd);
    const float4* ew = (const float4*)(EFW + (size_t)(i * Nn + j) * EHd);
    if (jj < 7) {  // pull next j-row toward the WGP (global_prefetch_b8)
      __builtin_prefetch(ZJ + (size_t)(b * Nn + j + 1) * EHd + lane * 8, 0, 3);
      __builtin_prefetch(EFW + (size_t)(i * Nn + j + 1) * EHd + lane * 8, 0, 3);
    }
    float acc = 0.0f;
#pragma unroll
    for (int c = 0; c < 2; ++c) {
      const int k4 = c * 32 + lane;
      const float4 zv = zj[k4], ev = ew[k4], iv = zi4[k4], wv = w24[k4];
      acc += fmaxf(iv.x + zv.x + ev.x, 0.0f) * wv.x;
      acc += fmaxf(iv.y + zv.y + ev.y, 0.0f) * wv.y;
      acc += fmaxf(iv.z + zv.z + ev.z, 0.0f) * wv.z;
      acc += fmaxf(iv.w + zv.w + ev.w, 0.0f) * wv.w;
    }
    acc = wave_sum32(acc);
    if (lane == 0) {
      const float p = sigmoidf_(acc + b2) * mask[i * Nn + j];
      out[(size_t)b * OUT_STRIDE + 80 + i * Nn + j] = p;
      if (i == 0) poe[b * Nn + j] = (j == 0) ? 0.0f : p;  // npm masks PRIMARY
    }
  }
}

// ---------------------------------------------------------------------------
// Fusion: susc_ctx / edge_ctx / diff, concat with trunk, LN2 -> ln2out[B,388]
// One block per batch, 128 threads.
// ---------------------------------------------------------------------------
__global__ __launch_bounds__(128)
void fuse_kernel(const float* __restrict__ z, const float* __restrict__ organ_s,
                 const float* __restrict__ poe, const float* __restrict__ trunk,
                 const float* __restrict__ g2, const float* __restrict__ b2,
                 float* __restrict__ ln2out, float* __restrict__ diffbuf)
{
  const int b = blockIdx.x;
  const int t = threadIdx.x;
  __shared__ float sn[Nn], pn[Nn];
  __shared__ float fused[FUSED_DIM];
  __shared__ float ssum_s, psum_s, smax_s, pmax_s, mu_s, var_s;

  if (t < Nn) {
    sn[t] = (t == 0) ? 0.0f : organ_s[b * Nn + t];  // susc_np (primary masked)
    pn[t] = poe[b * Nn + t];
  }
  __syncthreads();
  if (t == 0) {
    float ss = 0.0f, sm = -1e30f, ps = 0.0f, pm = -1e30f;
    for (int n = 0; n < Nn; ++n) {
      ss += sn[n]; sm = fmaxf(sm, sn[n]);
      ps += pn[n]; pm = fmaxf(pm, pn[n]);
    }
    ssum_s = ss; smax_s = sm; psum_s = ps; pmax_s = pm;
  }
  __syncthreads();
  const float sden = fmaxf(ssum_s, 1e-6f);
  const float pden = fmaxf(psum_s, 1e-6f);
  {
    float scx = 0.0f, ecx = 0.0f;
    for (int n = 0; n < Nn; ++n) {
      const float zv = z[(size_t)(b * Nn + n) * Dd + t];
      scx += sn[n] * zv;
      ecx += pn[n] * zv;
    }
    fused[t]            = trunk[(size_t)b * THd + t];
    fused[THd + t]      = scx / sden;
    fused[THd + Dd + t] = ecx / pden;
  }
  if (t < 4) {
    const float dv = (t == 0) ? ssum_s / Nn
                   : (t == 1) ? smax_s
                   : (t == 2) ? psum_s / Nn
                              : pmax_s;
    fused[THd + 2 * Dd + t] = dv;
    diffbuf[b * 4 + t] = dv;
  }
  __syncthreads();
  if (t == 0) {
    float mu = 0.0f;
    for (int d = 0; d < FUSED_DIM; ++d) mu += fused[d];
    mu /= FUSED_DIM;
    float var = 0.0f;
    for (int d = 0; d < FUSED_DIM; ++d) { float c = fused[d] - mu; var += c * c; }
    var /= FUSED_DIM;
    mu_s = mu; var_s = var;
  }
  __syncthreads();
  const float rs = rsqrtf(var_s + 1e-5f);
  for (int idx = t; idx < FUSED_DIM; idx += 128)
    ln2out[(size_t)b * FUSED_DIM + idx] =
        (fused[idx] - mu_s) * rs * g2[idx] + b2[idx];
}

// ---------------------------------------------------------------------------
// Final heads. One wave per batch; lanes split across the small dot products.
// Output per batch row: [0:8] hazard, [8] rec_logit, [9:12] rec_loc,
// [12] expl_rec, [13:16] expl_loc.
// ---------------------------------------------------------------------------
__global__ __launch_bounds__(32)
void heads_kernel(const float* __restrict__ joint, const float* __restrict__ diffbuf,
                  const float* __restrict__ organ_s, const float* __restrict__ poe,
                  const float* __restrict__ Wos, const float* __restrict__ bos,
                  const float* __restrict__ Wr, const float* __restrict__ br,
                  const float* __restrict__ Wl, const float* __restrict__ bl,
                  const float* __restrict__ Wx1, const float* __restrict__ bx1,
                  const float* __restrict__ Wx2, const float* __restrict__ bx2,
                  const float* __restrict__ bucket, float* __restrict__ out)
{
  const int b = blockIdx.x;
  const int t = threadIdx.x;
  const float* jv = joint + (size_t)b * THd;
  float* ob = out + (size_t)b * OUT_STRIDE;
  if (t < 8) {                       // hazard
    float a = 0.0f;
    for (int k = 0; k < THd; ++k) a += jv[k] * Wos[k * 8 + t];
    ob[t] = a + bos[t];
  } else if (t == 8) {               // rec_logit
    float a = 0.0f;
    for (int k = 0; k < THd; ++k) a += jv[k] * Wr[k];
    ob[8] = a + br[0];
  } else if (t < 12) {               // rec_loc
    const int j = t - 9;
    float a = 0.0f;
    for (int k = 0; k < THd; ++k) a += jv[k] * Wl[k * 3 + j];
    ob[9 + j] = a + bl[j];
  } else if (t == 12) {              // expl_rec: relu(diff@Wx1+bx1)@Wx2+bx2
    const float* dv = diffbuf + b * 4;
    float a = 0.0f;
    for (int h = 0; h < 16; ++h) {
      float hh = bx1[h];
      for (int k = 0; k < 4; ++k) hh += dv[k] * Wx1[k * 16 + h];
      a += fmaxf(hh, 0.0f) * Wx2[h];
    }
    ob[12] = a + bx2[0];
  } else if (t < 16) {               // expl_loc = logit(clip(loc_scores))
    const int j = t - 13;
    float a = 0.0f;
    for (int n = 0; n < Nn; ++n)
      a += (0.5f * organ_s[b * Nn + n] + 0.5f * poe[b * Nn + n]) * bucket[j * Nn + n];
    const float ls = fminf(fmaxf(a, 1e-4f), 1.0f - 1e-4f);
    ob[13 + j] = logf(ls) - log1pf(-ls);
  }
}

// ---------------------------------------------------------------------------
extern "C" void kernel_launch(void* const* d_in, const int* in_sizes, int n_in,
                              void* d_out, int out_size, void* d_ws, size_t ws_size,
                              hipStream_t stream) {
  (void)in_sizes; (void)n_in; (void)out_size; (void)ws_size;
  // Inputs in setup_inputs() insertion order; params dict flattened in order.
  const float* z    = (const float*)d_in[0];   // [128,64,128]
  const float* ef   = (const float*)d_in[1];   // [64,64,8]
  const float* mask = (const float*)d_in[2];   // [64,64]
  const float* bkt  = (const float*)d_in[3];   // [3,64]
  const float* Wp1  = (const float*)d_in[4];
  const float* bp1  = (const float*)d_in[5];
  const float* Wp2  = (const float*)d_in[6];
  const float* bp2  = (const float*)d_in[7];
  const float* g1   = (const float*)d_in[8];
  const float* b1   = (const float*)d_in[9];
  const float* Wt   = (const float*)d_in[10];
  const float* bt   = (const float*)d_in[11];
  const float* Wo1  = (const float*)d_in[12];
  const float* bo1  = (const float*)d_in[13];
  const float* Wo2  = (const float*)d_in[14];
  const float* bo2  = (const float*)d_in[15];
  const float* We1  = (const float*)d_in[16];  // [264,256]: Wi|Wj|We
  const float* be1  = (const float*)d_in[17];
  const float* We2  = (const float*)d_in[18];
  const float* be2  = (const float*)d_in[19];
  const float* g2   = (const float*)d_in[20];
  const float* b2   = (const float*)d_in[21];
  const float* Wf   = (const float*)d_in[22];
  const float* bf   = (const float*)d_in[23];
  const float* Wos  = (const float*)d_in[24];
  const float* bos  = (const float*)d_in[25];
  const float* Wr   = (const float*)d_in[26];
  const float* br   = (const float*)d_in[27];
  const float* Wl   = (const float*)d_in[28];
  const float* bl   = (const float*)d_in[29];
  const float* Wx1  = (const float*)d_in[30];
  const float* bx1  = (const float*)d_in[31];
  const float* Wx2  = (const float*)d_in[32];
  const float* bx2  = (const float*)d_in[33];
  float* out = (float*)d_out;

  // Workspace layout (floats), ~34 MB total
  float* ws = (float*)d_ws;
  const int MZ = Bb * Nn;                    // 8192 z-rows
  float* S1      = ws;                       // 8192*128
  float* O1      = S1  + (size_t)MZ * Dd;    // 8192*256
  float* ZI      = O1  + (size_t)MZ * EHd;   // 8192*256
  float* ZJ      = ZI  + (size_t)MZ * EHd;   // 8192*256
  float* EFW     = ZJ  + (size_t)MZ * EHd;   // 4096*256
  float* organ_s = EFW + (size_t)Nn * Nn * EHd;  // 8192
  float* poe     = organ_s + MZ;             // 8192
  float* lnpool  = poe + Bb * Nn;            // 128*128
  float* trunk   = lnpool + Bb * Dd;         // 128*128
  float* ln2out  = trunk + Bb * THd;         // 128*388
  float* joint   = ln2out + (size_t)Bb * FUSED_DIM;  // 128*128
  float* diffb   = joint + Bb * THd;         // 128*4

  dim3 blk(128);
  // 1) S1 = tanh(z @ Wp1 + bp1)                 [8192,128]
  gemm_wmma_f32<<<dim3(MZ/16, Dd/64), blk, 0, stream>>>(z, Dd, Wp1, Dd, bp1, S1, Dd, Dd, 2);
  // 2) O1 = relu(z @ Wo1 + bo1)                 [8192,256]
  gemm_wmma_f32<<<dim3(MZ/16, EHd/64), blk, 0, stream>>>(z, Dd, Wo1, EHd, bo1, O1, EHd, Dd, 1);
  // 3) ZI = z @ Wi   (We1 rows 0..127)          [8192,256]
  gemm_wmma_f32<<<dim3(MZ/16, EHd/64), blk, 0, stream>>>(z, Dd, We1, EHd, nullptr, ZI, EHd, Dd, 0);
  // 4) ZJ = z @ Wj   (We1 rows 128..255)
  gemm_wmma_f32<<<dim3(MZ/16, EHd/64), blk, 0, stream>>>(z, Dd, We1 + 128 * EHd, EHd, nullptr, ZJ, EHd, Dd, 0);
  // 5) EFW = ef @ We (We1 rows 256..263)        [4096,256], K=8
  gemm_wmma_f32<<<dim3((Nn*Nn)/16, EHd/64), blk, 0, stream>>>(ef, EFd, We1 + 256 * EHd, EHd, nullptr, EFW, EHd, EFd, 0);
  // 6) organ_susc
  organ_kernel<<<dim3(MZ/8), dim3(256), 0, stream>>>(O1, Wo2, bo2, organ_s, out);
  // 7) attention pool + LN1
  pool_kernel<<<dim3(Bb), dim3(256), 0, stream>>>(z, S1, Wp2, bp2, g1, b1, lnpool);
  // 8) trunk = relu(lnpool @ Wt + bt)           [128,128]
  gemm_wmma_f32<<<dim3(Bb/16, THd/64), blk, 0, stream>>>(lnpool, Dd, Wt, THd, bt, trunk, THd, Dd, 1);
  // 9) edge head (also fills output edge section + poe)
  edge_kernel<<<dim3(Bb * Nn), dim3(256), 0, stream>>>(ZI, ZJ, EFW, be1, We2, be2, mask, out, poe);
  // 10) fusion + LN2
  fuse_kernel<<<dim3(Bb), dim3(128), 0, stream>>>(z, organ_s, poe, trunk, g2, b2, ln2out, diffb);
  // 11) joint = relu(ln2out @ Wf + bf)          [128,128], K=388
  gemm_wmma_f32<<<dim3(Bb/16, THd/64), blk, 0, stream>>>(ln2out, FUSED_DIM, Wf, THd, bf, joint, THd, FUSED_DIM, 1);
  // 12) heads
  heads_kernel<<<dim3(Bb), dim3(32), 0, stream>>>(joint, diffb, organ_s, poe,
                                                  Wos, bos, Wr, br, Wl, bl,
                                                  Wx1, bx1, Wx2, bx2, bkt, out);
}